// VectorQuantizerEMA_10857677325067
// MI455X (gfx1250) — compile-verified
//
#include <hip/hip_runtime.h>
#include <hip/hip_bf16.h>
#include <math.h>

// ---------------- problem constants ----------------
#define NUM_EMB 512
#define EMB_DIM 64
#define NN      131072          // 16 * 8192 rows
#define LOG2_NN 17
#define COMMIT  0.25f
#define DECAY   0.99f
#define ONE_MINUS_DECAY 0.01f
#define EPSV    1e-5f

// output layout (floats): loss | quantized [D*N] | perplexity | encodings [N*K] | distances [N*K]
#define Q_OFF    ((size_t)1)
#define PERP_OFF ((size_t)(1 + (size_t)EMB_DIM * NN))
#define ENC_OFF  ((size_t)(PERP_OFF + 1))
#define DIST_OFF ((size_t)(ENC_OFF + (size_t)NN * NUM_EMB))

typedef __attribute__((ext_vector_type(2))) float v2f;
typedef __attribute__((ext_vector_type(8))) float v8f;
typedef __attribute__((address_space(3))) float lds_f32;

// ---------------- kernel 1: init ----------------
// 1 block x 512 threads: colnorm, zero counts/dw/loss accumulator
__global__ void vq_init(const float* __restrict__ emb,
                        float* __restrict__ counts,
                        float* __restrict__ dw,
                        float* __restrict__ colnorm,
                        float* __restrict__ lossacc) {
  int k = threadIdx.x;           // 0..511
  float s = 0.f;
#pragma unroll 8
  for (int d = 0; d < EMB_DIM; ++d) {
    float e = emb[k * EMB_DIM + d];
    s += e * e;
  }
  colnorm[k] = s;
  counts[k] = 0.f;
#pragma unroll 4
  for (int d = 0; d < EMB_DIM; ++d) dw[k * EMB_DIM + d] = 0.f;
  if (k == 0) *lossacc = 0.f;
}

// ---------------- kernel 2: distances + argmin (WMMA fp32) ----------------
// grid = NN/128 blocks of 256 threads (8 waves). Wave w handles rows
// [blockIdx*128 + w*16, +16). Embedding staged in LDS (128 KB) via
// CDNA5 async global->LDS DMA (ASYNCcnt path).
__global__ __launch_bounds__(256) void vq_dist(
    const float* __restrict__ inp,       // [D, N] == flat^T
    const float* __restrict__ emb,       // [K, D]
    const float* __restrict__ colnorm,   // [K]
    float* __restrict__ dist,            // [N, K]
    float* __restrict__ enc,             // [N, K] (zero-filled here)
    int*   __restrict__ idx,             // [N]
    float* __restrict__ counts)          // [K]
{
  extern __shared__ float semb[];        // NUM_EMB * EMB_DIM floats
  const int tid = threadIdx.x;

  // hint the A-slab into cache while we stage B
  __builtin_prefetch(&inp[blockIdx.x * 128], 0, 0);

  // --- stage embedding into LDS with async DMA (global_load_async_to_lds_b128)
  // AS(3) pointers are 32-bit LDS byte offsets on amdgcn.
  {
    const unsigned lds_base = (unsigned)(uintptr_t)(lds_f32*)semb;
    for (int i = tid; i < NUM_EMB * EMB_DIM / 4; i += 256) {
      const unsigned goff = (unsigned)i * 16u;      // byte offset into emb
      const unsigned laddr = lds_base + goff;       // LDS byte address
      asm volatile("global_load_async_to_lds_b128 %0, %1, %2"
                   :
                   : "v"(laddr), "v"(goff), "s"(emb)
                   : "memory");
    }
    asm volatile("s_wait_asynccnt 0x0" ::: "memory");
  }
  __syncthreads();

  const int lane = tid & 31;
  const int wave = tid >> 5;
  const int hi   = lane >> 4;            // 0 | 1 (half-wave)
  const int lm   = lane & 15;
  const int row0 = blockIdx.x * 128 + wave * 16;

  // --- A fragments: 16 steps of K=4 over D=64. ISA layout:
  // VGPR0: lanes0-15 -> (M=lm, K=4s+0), lanes16-31 -> (M=lm, K=4s+2)
  // VGPR1: K=4s+1 / 4s+3. flat[n][k] = inp[k*N + n].
  v2f afrag[16];
  float psum = 0.f;
#pragma unroll
  for (int s = 0; s < 16; ++s) {
    const int k0 = 4 * s + 2 * hi;
    float x0 = inp[(size_t)k0 * NN + row0 + lm];
    float x1 = inp[(size_t)(k0 + 1) * NN + row0 + lm];
    afrag[s].x = x0; afrag[s].y = x1;
    psum += x0 * x0 + x1 * x1;
  }
  // row norms: each half-wave summed half the k's
  float rn = psum + __shfl_xor(psum, 16, 32);   // lane L holds ||row row0+lm||^2
  float rnv[8];
#pragma unroll
  for (int j = 0; j < 8; ++j) rnv[j] = __shfl(rn, j + 8 * hi, 32); // C-layout row of VGPR j

  float minv[8];
  int   minc[8];
#pragma unroll
  for (int j = 0; j < 8; ++j) { minv[j] = INFINITY; minc[j] = 0; }

  // --- loop over 32 column tiles of 16 codes
  for (int ct = 0; ct < 32; ++ct) {
    const int col0 = ct * 16;
    v8f acc = {};
#pragma unroll
    for (int s = 0; s < 16; ++s) {
      // B (4x16) fragment, mirror of A layout: v0 -> K=4s+2*hi, v1 -> +1,
      // column = col0+lm. B[k][c] = emb[c][k]  (consecutive k -> ds_load_b64)
      const float* bp = &semb[(col0 + lm) * EMB_DIM + 4 * s + 2 * hi];
      v2f b; b.x = bp[0]; b.y = bp[1];
      acc = __builtin_amdgcn_wmma_f32_16x16x4_f32(
          false, afrag[s], false, b, (short)0, acc, false, false);
    }
    const float cn = colnorm[col0 + lm];
#pragma unroll
    for (int j = 0; j < 8; ++j) {
      // C layout: lane L, VGPR j -> (M = j + 8*hi, N = lm)
      const float dv = rnv[j] + cn - 2.0f * acc[j];
      const int row = row0 + j + 8 * hi;
      dist[(size_t)row * NUM_EMB + col0 + lm] = dv;
      if (dv < minv[j]) { minv[j] = dv; minc[j] = col0 + lm; }
    }
  }

  // --- argmin reduce across the 16 lanes holding each row (first-index ties)
#pragma unroll
  for (int j = 0; j < 8; ++j) {
#pragma unroll
    for (int off = 8; off >= 1; off >>= 1) {
      float ov = __shfl_xor(minv[j], off, 32);
      int   oc = __shfl_xor(minc[j], off, 32);
      if (ov < minv[j] || (ov == minv[j] && oc < minc[j])) { minv[j] = ov; minc[j] = oc; }
    }
  }
  if (lm == 0) {
#pragma unroll
    for (int j = 0; j < 8; ++j) {
      const int row = row0 + j + 8 * hi;
      idx[row] = minc[j];
      atomicAdd(&counts[minc[j]], 1.0f);
    }
  }

  // --- zero this wave's encodings block (ones scattered later)
  {
    float4 z = make_float4(0.f, 0.f, 0.f, 0.f);
    float4* encb = (float4*)&enc[(size_t)row0 * NUM_EMB];
    for (int t = lane; t < 16 * NUM_EMB / 4; t += 32) encb[t] = z;
  }
}

// ---------------- kernel 3: dw segment-sum via fp32 atomics ----------------
// grid covers D*N elements; g = d*N + n -> coalesced input & idx reads
__global__ __launch_bounds__(256) void vq_dw(const float* __restrict__ inp,
                                             const int* __restrict__ idx,
                                             float* __restrict__ dw) {
  const size_t g = (size_t)blockIdx.x * 256 + threadIdx.x;
  const int d = (int)(g >> LOG2_NN);
  const int n = (int)(g & (NN - 1));
  atomicAdd(&dw[idx[n] * EMB_DIM + d], inp[g]);
}

// ---------------- kernel 4: EMA codebook update + perplexity ----------------
__global__ __launch_bounds__(512) void vq_finalize(
    const float* __restrict__ ema_cs, const float* __restrict__ ema_w,
    const float* __restrict__ counts, const float* __restrict__ dw,
    float* __restrict__ new_emb, float* __restrict__ perp_out) {
  __shared__ float red[NUM_EMB];
  __shared__ float clf[NUM_EMB];
  const int k = threadIdx.x;

  const float cluster = ema_cs[k] * DECAY + ONE_MINUS_DECAY * counts[k];
  red[k] = cluster;
  __syncthreads();
  for (int off = 256; off > 0; off >>= 1) {
    if (k < off) red[k] += red[k + off];
    __syncthreads();
  }
  const float nsum = red[0];
  __syncthreads();
  clf[k] = (cluster + EPSV) / (nsum + NUM_EMB * EPSV) * nsum;

  // perplexity
  const float p = counts[k] * (1.0f / (float)NN);
  red[k] = -p * logf(p + 1e-10f);
  __syncthreads();
  for (int off = 256; off > 0; off >>= 1) {
    if (k < off) red[k] += red[k + off];
    __syncthreads();
  }
  if (k == 0) *perp_out = expf(red[0]);

  // new embedding
  for (int e = k; e < NUM_EMB * EMB_DIM; e += 512)
    new_emb[e] = (ema_w[e] * DECAY + ONE_MINUS_DECAY * dw[e]) / clf[e >> 6];
}

// ---------------- kernel 5: quantize gather + loss + encodings ones -------
__global__ __launch_bounds__(256) void vq_quant(
    const float* __restrict__ inp, const int* __restrict__ idx,
    const float* __restrict__ new_emb,
    float* __restrict__ q_out,          // [D, N]
    float* __restrict__ enc,            // [N, K]
    float* __restrict__ lossacc) {
  __shared__ float red[256];
  const size_t g = (size_t)blockIdx.x * 256 + threadIdx.x;
  const int d = (int)(g >> LOG2_NN);
  const int n = (int)(g & (NN - 1));
  const int c = idx[n];
  const float q = new_emb[c * EMB_DIM + d];
  q_out[g] = q;                          // quantized_st == quantized in value
  if (d == 0) enc[(size_t)n * NUM_EMB + c] = 1.0f;
  const float diff = q - inp[g];
  red[threadIdx.x] = diff * diff;
  __syncthreads();
  for (int off = 128; off > 0; off >>= 1) {
    if (threadIdx.x < off) red[threadIdx.x] += red[threadIdx.x + off];
    __syncthreads();
  }
  if (threadIdx.x == 0) atomicAdd(lossacc, red[0]);
}

// ---------------- kernel 6: scalar loss ----------------
__global__ void vq_loss(const float* __restrict__ lossacc, float* __restrict__ out) {
  out[0] = COMMIT * (*lossacc) / (float)((size_t)NN * EMB_DIM);
}

// ---------------- launch ----------------
extern "C" void kernel_launch(void* const* d_in, const int* in_sizes, int n_in,
                              void* d_out, int out_size, void* d_ws, size_t ws_size,
                              hipStream_t stream) {
  const float* inp    = (const float*)d_in[0];   // [64, 16, 8192] = [D, N]
  const float* emb    = (const float*)d_in[1];   // [512, 64]
  const float* ema_w  = (const float*)d_in[2];   // [512, 64]
  const float* ema_cs = (const float*)d_in[3];   // [512]
  float* out = (float*)d_out;

  // workspace carve-up (all 16B aligned)
  char* ws = (char*)d_ws;
  int*   idx     = (int*)ws;                         ws += (size_t)NN * sizeof(int);      // 512 KB
  float* counts  = (float*)ws;                       ws += NUM_EMB * sizeof(float);
  float* dw      = (float*)ws;                       ws += NUM_EMB * EMB_DIM * sizeof(float);
  float* colnorm = (float*)ws;                       ws += NUM_EMB * sizeof(float);
  float* new_emb = (float*)ws;                       ws += NUM_EMB * EMB_DIM * sizeof(float);
  float* lossacc = (float*)ws;                       ws += 16;

  float* q_out = out + Q_OFF;
  float* perp  = out + PERP_OFF;
  float* enc   = out + ENC_OFF;
  float* dist  = out + DIST_OFF;

  vq_init<<<1, NUM_EMB, 0, stream>>>(emb, counts, dw, colnorm, lossacc);

  const size_t lds_bytes = (size_t)NUM_EMB * EMB_DIM * sizeof(float); // 128 KB
  vq_dist<<<NN / 128, 256, lds_bytes, stream>>>(inp, emb, colnorm, dist, enc, idx, counts);

  vq_dw<<<(NN * EMB_DIM) / 256, 256, 0, stream>>>(inp, idx, dw);

  vq_finalize<<<1, NUM_EMB, 0, stream>>>(ema_cs, ema_w, counts, dw, new_emb, perp);

  vq_quant<<<(NN * EMB_DIM) / 256, 256, 0, stream>>>(inp, idx, new_emb, q_out, enc, lossacc);

  vq_loss<<<1, 1, 0, stream>>>(lossacc, out);
}